// ann_deep_44641890075304
// MI455X (gfx1250) — compile-verified
//
#include <hip/hip_runtime.h>
#include <math.h>

typedef __attribute__((ext_vector_type(2))) float v2f;
typedef __attribute__((ext_vector_type(8))) float v8f;
typedef __attribute__((ext_vector_type(4))) unsigned int v4u;
typedef __attribute__((ext_vector_type(8))) int v8i_t;
typedef __attribute__((ext_vector_type(4))) int v4i_t;

#define Bdim 512
#define Ndim 8192
#define Kdim 16
#define Hdim 16

#if defined(__gfx1250__) && __has_builtin(__builtin_amdgcn_tensor_load_to_lds)
#define USE_TDM_GATHER 1
#else
#define USE_TDM_GATHER 0
#endif

// ---------------------------------------------------------------------------
// Tiled transpose: in (R, C) row-major  ->  out (C, R) row-major
// ---------------------------------------------------------------------------
__global__ __launch_bounds__(256) void transpose_tile(const float* __restrict__ in,
                                                      float* __restrict__ out,
                                                      int R, int C) {
  __shared__ float tile[32][33];
  const int cb = blockIdx.x * 32;
  const int rb = blockIdx.y * 32;
  const int tx = threadIdx.x, ty = threadIdx.y;
#pragma unroll
  for (int i = 0; i < 32; i += 8)
    tile[ty + i][tx] = in[(size_t)(rb + ty + i) * C + (cb + tx)];
  __syncthreads();
#pragma unroll
  for (int i = 0; i < 32; i += 8)
    out[(size_t)(cb + ty + i) * R + (rb + tx)] = tile[tx][ty + i];
}

// ---------------------------------------------------------------------------
// Per-node fused kernel: one block (4 wave32s) per node n.
//  - TDM gather-mode DMA pulls the 16 indexed xT rows (32KB) into LDS
//  - each wave computes 8 tiles of 16 batch rows with v_wmma_f32_16x16x4_f32
//  - packed shfl_xor butterfly (15 shuffles) for the H @ W2 contraction,
//    results land one-per-lane -> coalesced 64B store per tile
// ---------------------------------------------------------------------------
__global__ __launch_bounds__(128) void gnn_node_kernel(
    const float* __restrict__ xT,      // (N, B)
    const int* __restrict__ neighs,    // (N, K)
    const float* __restrict__ W1,      // (N, K, H)
    const float* __restrict__ b1,      // (N, H)
    const float* __restrict__ W2,      // (N, H)
    const float* __restrict__ b2,      // (N,)
    float* __restrict__ outT) {        // (N, B)
  __shared__ float Gt[Kdim * Bdim];    // gathered rows, k-major: 32KB
  __shared__ float W1s[Kdim * Hdim];   // 16x16 weights, k-major
  __shared__ float b1s[Hdim];
  __shared__ float W2s[Hdim];

  const int n = blockIdx.x;
  const int tid = threadIdx.x;
  const int lane = tid & 31;
  const int wave = tid >> 5;

#if USE_TDM_GATHER
  if (wave == 0) {
    // Tensor DMA, gather mode: 16 rows (indices = neighs[n,:]) of the
    // 8192x512 f32 tensor xT -> LDS Gt, rows stored consecutively (k-major).
    const int* nb = neighs + n * Kdim;
    const uint32_t lds_off = (uint32_t)(uintptr_t)(void*)&Gt[0];
    const uint64_t gaddr = (uint64_t)(uintptr_t)xT;

    // D# group 0: count=1 | gather_mode=1 (16-bit idx), lds_addr, global_addr,
    //             type=2 ("image") in bits 127:126.
    v4u g0 = {0x80000001u, lds_off, (uint32_t)gaddr,
              (uint32_t)((gaddr >> 32) & 0x01FFFFFFu) | 0x80000000u};
    // D# group 1: data_size=2 (4B) @17:16 | tensor_dim0=512 @79:48 |
    //             tensor_dim1=8192 @111:80 | tile_dim0=512 @127:112 |
    //             tile_dim1=16 valid indices @143:128 | dim0_stride=512 @207:160
    v8i_t g1 = {(int)0x00020000, (int)(512u << 16), (int)(8192u << 16),
                (int)(512u << 16), 16, 512, 0, 0};
    // D# groups 2/3: sixteen 16-bit row indices, 2 per dword.
    v4i_t g2 = {(int)(((uint32_t)nb[1] << 16) | (uint32_t)nb[0]),
                (int)(((uint32_t)nb[3] << 16) | (uint32_t)nb[2]),
                (int)(((uint32_t)nb[5] << 16) | (uint32_t)nb[4]),
                (int)(((uint32_t)nb[7] << 16) | (uint32_t)nb[6])};
    v4i_t g3 = {(int)(((uint32_t)nb[9] << 16) | (uint32_t)nb[8]),
                (int)(((uint32_t)nb[11] << 16) | (uint32_t)nb[10]),
                (int)(((uint32_t)nb[13] << 16) | (uint32_t)nb[12]),
                (int)(((uint32_t)nb[15] << 16) | (uint32_t)nb[14])};
    // 6-arg toolchain: 5th operand maps to the encoding's unused VADDR4 -> 0.
    v8i_t g4 = {0, 0, 0, 0, 0, 0, 0, 0};
    __builtin_amdgcn_tensor_load_to_lds(g0, g1, g2, g3, g4, 0);
    __builtin_amdgcn_s_wait_tensorcnt(0);
  }
#else
  {
    const float4* xT4 = (const float4*)xT;
    float4* Gt4 = (float4*)Gt;
#pragma unroll
    for (int r = 0; r < Kdim; ++r) {
      const int idx = neighs[n * Kdim + r];
      Gt4[r * (Bdim / 4) + tid] = xT4[(size_t)idx * (Bdim / 4) + tid];
    }
  }
#endif
  if (tid < 64) {  // W1[n]: 256 floats = 64 float4
    ((float4*)W1s)[tid] = ((const float4*)W1)[(size_t)n * 64 + tid];
  }
  if (tid < Hdim) {
    b1s[tid] = b1[n * Hdim + tid];
    W2s[tid] = W2[n * Hdim + tid];
  }
  __syncthreads();

  const float b2v = b2[n];
  const int hi = lane >> 4;  // half-wave: selects K pair for A/B, M+8 for C/D
  const int lc = lane & 15;  // M for A, h-column for B/C/D

  // Hoist the half-wave K offset into the base pointers so the per-chunk LDS
  // offsets are compile-time constants -> ds_load_2addr_stride64 candidates.
  const float* __restrict__ wbase = &W1s[2 * hi * Hdim + lc];

  for (int t = wave; t < Bdim / 16; t += 4) {
    const int b0 = t * 16;
    const float* __restrict__ gbase = &Gt[2 * hi * Bdim + b0 + lc];
    v8f acc = {};
#pragma unroll
    for (int j = 0; j < 4; ++j) {
      // A rows K = 4j+2hi, 4j+2hi+1 at constant offsets from gbase
      v2f a, bm;
      a.x = gbase[(4 * j) * Bdim];
      a.y = gbase[(4 * j + 1) * Bdim];
      bm.x = wbase[(4 * j) * Hdim];
      bm.y = wbase[(4 * j + 1) * Hdim];
      acc = __builtin_amdgcn_wmma_f32_16x16x4_f32(false, a, false, bm,
                                                  (short)0, acc, false, false);
    }
    // Epilogue: D[v] holds (M = b0 + v + 8*hi, h = lc).
    const float b1v = b1s[lc];
    const float w2v = W2s[lc];
    float tt[8];
#pragma unroll
    for (int v = 0; v < 8; ++v) {
      float h = acc[v] + b1v;
      h = h > 0.0f ? h : 0.0f;
      tt[v] = h * w2v;
    }
    // Packed butterfly reduction over the 16 h-lanes of each half-wave:
    // fold the register set in half after every xor round (15 shuffles total).
#pragma unroll
    for (int v = 0; v < 8; ++v) tt[v] += __shfl_xor(tt[v], 1);
    float p0 = (lane & 1) ? tt[1] : tt[0];
    float p1 = (lane & 1) ? tt[3] : tt[2];
    float p2 = (lane & 1) ? tt[5] : tt[4];
    float p3 = (lane & 1) ? tt[7] : tt[6];
    p0 += __shfl_xor(p0, 2);
    p1 += __shfl_xor(p1, 2);
    p2 += __shfl_xor(p2, 2);
    p3 += __shfl_xor(p3, 2);
    float q0 = (lane & 2) ? p1 : p0;
    float q1 = (lane & 2) ? p3 : p2;
    q0 += __shfl_xor(q0, 4);
    q1 += __shfl_xor(q1, 4);
    float r = (lane & 4) ? q1 : q0;
    r += __shfl_xor(r, 8);
    // lane l now holds S[l&7] for M = b0 + 8*hi + (l&7); lanes 0-7 and 16-23
    // store 16 consecutive floats -> one coalesced 64B store.
    if (!(lane & 8)) {
      const float z = r + b2v;
      outT[(size_t)n * Bdim + b0 + hi * 8 + (lane & 7)] =
          1.0f / (1.0f + __expf(-z));
    }
  }
}

extern "C" void kernel_launch(void* const* d_in, const int* in_sizes, int n_in,
                              void* d_out, int out_size, void* d_ws, size_t ws_size,
                              hipStream_t stream) {
  const float* x = (const float*)d_in[0];        // (B, N)
  const int* neighs = (const int*)d_in[1];       // (N, K)
  const float* W1 = (const float*)d_in[2];       // (N, K, H)
  const float* b1 = (const float*)d_in[3];       // (N, H)
  const float* W2 = (const float*)d_in[4];       // (N, H)
  const float* b2 = (const float*)d_in[5];       // (N,)
  float* out = (float*)d_out;                    // (B, N)

  float* xT = (float*)d_ws;                      // 16 MB: (N, B)
  float* outT = xT + (size_t)Ndim * Bdim;        // 16 MB: (N, B)

  // x (B,N) -> xT (N,B): every gather becomes a contiguous 2KB row read
  transpose_tile<<<dim3(Ndim / 32, Bdim / 32), dim3(32, 8), 0, stream>>>(
      x, xT, Bdim, Ndim);

  gnn_node_kernel<<<Ndim, 128, 0, stream>>>(xT, neighs, W1, b1, W2, b2, outT);

  // outT (N,B) -> out (B,N)
  transpose_tile<<<dim3(Bdim / 32, Ndim / 32), dim3(32, 8), 0, stream>>>(
      outT, out, Ndim, Bdim);
}